// BlurModel_80187039416783
// MI455X (gfx1250) — compile-verified
//
#include <hip/hip_runtime.h>
#include <hip/hip_bf16.h>
#include <stdint.h>

// Problem geometry (from reference): x is (8,3,1024,1024) f32 -> 24 planes.
#define WIDTH   1024
#define HEIGHT  1024
#define KS      100
#define OW      925      // 1024 - 99
#define OH      925
#define NIMG    24
#define THRESH  0.129f
#define KVAL    1e-4f
#define RSEG    128      // output rows per vertical segment

#if __has_builtin(__builtin_amdgcn_global_load_async_to_lds_b128)
#define HAS_ASYNC_LDS 1
#else
#define HAS_ASYNC_LDS 0
#endif

// ---------------------------------------------------------------------------
// Kernel 1: horizontal sliding-window sum (width 100) per scanline.
// One 256-thread block per row. Row -> LDS (async copy on gfx1250), block
// prefix sum in LDS, windowed difference, coalesced store of 925 sums.
// ---------------------------------------------------------------------------
__global__ __launch_bounds__(256) void hblur_kernel(const float* __restrict__ x,
                                                    float* __restrict__ hb) {
    __shared__ float P[WIDTH];      // inclusive prefix sums of the row
    __shared__ float part[256];     // per-thread chunk partials
#if HAS_ASYNC_LDS
    __shared__ float raw[WIDTH];    // async-staged raw row
#endif

    const int  tid = threadIdx.x;
    const long row = blockIdx.x;                 // scanline index within chunk
    const float* src = x + row * WIDTH + tid * 4;

    float4 v;
#if HAS_ASYNC_LDS
    {
        typedef int v4i __attribute__((ext_vector_type(4)));
        typedef __attribute__((address_space(1))) v4i* gv4i_p;
        typedef __attribute__((address_space(3))) v4i* lv4i_p;

        v4i* gflat = (v4i*)const_cast<float*>(src);     // reinterpret pointee
        v4i* lflat = (v4i*)(&raw[tid * 4]);
        gv4i_p g = (gv4i_p)gflat;                        // generic -> AS(1)
        lv4i_p l = (lv4i_p)lflat;                        // generic -> AS(3)

        // 16B per lane, 512B per wave, 4KB per block: one async b128 each.
        __builtin_amdgcn_global_load_async_to_lds_b128(g, l, 0, 0);
#if __has_builtin(__builtin_amdgcn_s_wait_asynccnt)
        __builtin_amdgcn_s_wait_asynccnt(0);
#else
        asm volatile("s_wait_asynccnt 0" ::: "memory");
#endif
        __syncthreads();
        v = *(const float4*)(&raw[tid * 4]);
    }
#else
    v = *(const float4*)src;
#endif

    // serial inclusive scan of this thread's 4 elements
    const float l0 = v.x;
    const float l1 = l0 + v.y;
    const float l2 = l1 + v.z;
    const float l3 = l2 + v.w;
    part[tid] = l3;
    __syncthreads();

    // Hillis-Steele inclusive scan over 256 chunk sums
    #pragma unroll
    for (int off = 1; off < 256; off <<= 1) {
        float t = 0.0f;
        if (tid >= off) t = part[tid - off];
        __syncthreads();
        part[tid] += t;
        __syncthreads();
    }

    const float excl = part[tid] - l3;   // exclusive prefix of this chunk
    const int b = tid * 4;
    P[b + 0] = excl + l0;
    P[b + 1] = excl + l1;
    P[b + 2] = excl + l2;
    P[b + 3] = excl + l3;
    __syncthreads();

    // windowed difference: sum x[j..j+99] = P[j+99] - P[j-1]
    float* dst = hb + row * OW;
    #pragma unroll
    for (int j = tid; j < OW; j += 256) {
        const float s = P[j + KS - 1] - (j ? P[j - 1] : 0.0f);
        dst[j] = s;
    }
}

// ---------------------------------------------------------------------------
// Kernel 2: vertical sliding-window sum (width 100) + scale + threshold.
// Thread owns one column; running sum down a 128-row output segment.
// Lanes cover adjacent columns -> fully coalesced row reads/writes.
// ---------------------------------------------------------------------------
__global__ __launch_bounds__(256) void vblur_kernel(const float* __restrict__ hb,
                                                    float* __restrict__ out) {
    const int c   = blockIdx.x * 256 + threadIdx.x;
    const int img = blockIdx.z;
    const int r0  = blockIdx.y * RSEG;
    if (c >= OW) return;

    int rend = r0 + RSEG;
    if (rend > OH) rend = OH;

    const float* hp = hb + (long)img * HEIGHT * OW + c;
    float*       op = out + (long)img * OH * OW + c;

    float s = 0.0f;
    #pragma unroll 4
    for (int k = 0; k < KS; ++k)
        s += hp[(long)(r0 + k) * OW];

    for (int r = r0; r < rend; ++r) {
        if (r > r0)
            s += hp[(long)(r + KS - 1) * OW] - hp[(long)(r - 1) * OW];
        if (r + KS + 7 < HEIGHT)   // stream-prefetch the incoming row
            __builtin_prefetch(&hp[(long)(r + KS + 7) * OW], 0, 1);
        const float o = s * KVAL;
        op[(long)r * OW] = (o > THRESH) ? 1.0f : o;
    }
}

// ---------------------------------------------------------------------------
// Host launcher. Intermediate H (24 x 1024 x 925 f32 = 91 MB) lives in d_ws;
// if ws_size is smaller we chunk by image plane (3.8 MB each), serialized on
// `stream` (graph-capture safe: only kernel launches, no allocs/syncs).
// ---------------------------------------------------------------------------
extern "C" void kernel_launch(void* const* d_in, const int* in_sizes, int n_in,
                              void* d_out, int out_size, void* d_ws, size_t ws_size,
                              hipStream_t stream) {
    (void)in_sizes; (void)n_in; (void)out_size;

    const float* x  = (const float*)d_in[0];
    float*       out = (float*)d_out;
    float*       hb  = (float*)d_ws;

    const size_t per_img = (size_t)HEIGHT * OW * sizeof(float);
    int chunk = (int)(ws_size / per_img);
    if (chunk < 1)    chunk = 1;      // ws expected >= 4 MB
    if (chunk > NIMG) chunk = NIMG;

    for (int i0 = 0; i0 < NIMG; i0 += chunk) {
        const int n = (NIMG - i0 < chunk) ? (NIMG - i0) : chunk;

        dim3 g1((unsigned)(n * HEIGHT));
        hblur_kernel<<<g1, 256, 0, stream>>>(x + (size_t)i0 * HEIGHT * WIDTH, hb);

        dim3 g2((OW + 255) / 256, (OH + RSEG - 1) / RSEG, (unsigned)n);
        vblur_kernel<<<g2, 256, 0, stream>>>(hb, out + (size_t)i0 * (size_t)OH * OW);
    }
}